// FeatureTexel_13271448945253
// MI455X (gfx1250) — compile-verified
//
#include <hip/hip_runtime.h>

// FeatureTexel: bilinear gather from ft[100000,16,9,9] at 524288 (x, ind) samples.
// Memory-bound gather (~0.03 FLOP/B). Wave32 tile = 16 samples x 16 channels;
// blend done as diag(w_k) * V_k via 16x V_WMMA_F32_16X16X4_F32 per tile
// (matrix pipe is free under the gather latency). Output streamed with
// non-temporal stores so L2 stays dedicated to atlas reuse (~5.2x avg).

#define FD 16
#define FS 9

typedef float v2f __attribute__((ext_vector_type(2)));
typedef float v8f __attribute__((ext_vector_type(8)));

__global__ __launch_bounds__(256) void featex_wmma_kernel(
    const float* __restrict__ x,     // [nsamp, 2]
    const int*   __restrict__ inds,  // [nsamp]
    const float* __restrict__ ft,    // [NUM_ATLAS, FD, FS, FS]
    float*       __restrict__ out,   // [nsamp, FD]
    int nsamp)
{
    // Per-wave tile metadata: 16 samples x (4 corner offsets + 4 bilinear weights)
    __shared__ int   s_off[8][16][4];
    __shared__ float s_w[8][16][4];

    const int wave     = threadIdx.x >> 5;   // 8 waves per block
    const int lane     = threadIdx.x & 31;   // wave32
    const int tileBase = (blockIdx.x * 8 + wave) * 16;

    // ---- Phase 1: lanes 0-15 compute per-sample gather params ----
    if (lane < 16) {
        const int s   = tileBase + lane;
        const float xv  = x[2 * s + 0];
        const float yv  = x[2 * s + 1];
        const int   ind = inds[s];

        const float scale = 0.5f * (float)(FS - 1);
        float gx = fminf(fmaxf((xv + 1.0f) * scale, 0.0f), (float)(FS - 1));
        float gy = fminf(fmaxf((yv + 1.0f) * scale, 0.0f), (float)(FS - 1));
        const float x0f = floorf(gx);
        const float y0f = floorf(gy);
        const float wx  = gx - x0f;
        const float wy  = gy - y0f;
        const int x0 = (int)x0f;
        const int y0 = (int)y0f;
        const int x1 = (x0 + 1 < FS) ? (x0 + 1) : (FS - 1);
        const int y1 = (y0 + 1 < FS) ? (y0 + 1) : (FS - 1);

        const int base = ind * (FD * FS * FS);   // max ~1.3e8, fits int32
        s_off[wave][lane][0] = base + y0 * FS + x0;   // v00
        s_off[wave][lane][1] = base + y0 * FS + x1;   // v01
        s_off[wave][lane][2] = base + y1 * FS + x0;   // v10
        s_off[wave][lane][3] = base + y1 * FS + x1;   // v11

        s_w[wave][lane][0] = (1.0f - wx) * (1.0f - wy);
        s_w[wave][lane][1] = wx * (1.0f - wy);
        s_w[wave][lane][2] = (1.0f - wx) * wy;
        s_w[wave][lane][3] = wx * wy;
    }
    __syncthreads();

    // ---- Phase 2: gather B operands + 16x WMMA 16x16x4 (diag-blend) ----
    // Lane mapping: row (of diag A) == channel (column of B) == lane&15.
    // Lane half selects K half: A VGPR0/1 hold K={0,1} (lanes 0-15) or K={2,3};
    // B VGPR0/1 hold rows {4j,4j+1} (lanes 0-15) or {4j+2,4j+3} (lanes 16-31).
    const int row   = lane & 15;
    const int hsel  = (lane & 16) ? 2 : 0;
    const int chOff = row * (FS * FS);

    float wrow[4];
    #pragma unroll
    for (int k = 0; k < 4; ++k) wrow[k] = s_w[wave][row][k];

    int offA[4][4], offB[4][4];
    #pragma unroll
    for (int j = 0; j < 4; ++j) {
        const int sA = 4 * j + hsel;
        #pragma unroll
        for (int k = 0; k < 4; ++k) {
            offA[j][k] = s_off[wave][sA][k];
            offB[j][k] = s_off[wave][sA + 1][k];
        }
    }

    // Issue all 32 scattered gathers up front for maximum loads in flight.
    float bxv[16], byv[16];
    #pragma unroll
    for (int k = 0; k < 4; ++k) {
        #pragma unroll
        for (int j = 0; j < 4; ++j) {
            bxv[4 * k + j] = ft[offA[j][k] + chOff];
            byv[4 * k + j] = ft[offB[j][k] + chOff];
        }
    }

    v8f acc = {0.f, 0.f, 0.f, 0.f, 0.f, 0.f, 0.f, 0.f};
    #pragma unroll
    for (int k = 0; k < 4; ++k) {          // corner
        const float wk = wrow[k];
        #pragma unroll
        for (int j = 0; j < 4; ++j) {      // 4-row chunk of the 16x16 diagonal
            v2f a, b;
            b.x = bxv[4 * k + j];
            b.y = byv[4 * k + j];
            a.x = (row == 4 * j + hsel    ) ? wk : 0.0f;
            a.y = (row == 4 * j + hsel + 1) ? wk : 0.0f;
            // (neg_a, A, neg_b, B, c_mod, C, reuse_a, reuse_b)
            acc = __builtin_amdgcn_wmma_f32_16x16x4_f32(
                false, a, false, b, (short)0, acc, false, false);
        }
    }

    // ---- Phase 3: store D. VGPR r holds rows r (lanes 0-15) / r+8 (16-31),
    // column = lane&15 -> consecutive lanes hit consecutive channels (coalesced).
    const int rbase = tileBase + ((lane & 16) ? 8 : 0);
    #pragma unroll
    for (int r = 0; r < 8; ++r) {
        __builtin_nontemporal_store(acc[r], out + (size_t)(rbase + r) * FD + row);
    }
}

extern "C" void kernel_launch(void* const* d_in, const int* in_sizes, int n_in,
                              void* d_out, int out_size, void* d_ws, size_t ws_size,
                              hipStream_t stream) {
    const float* x    = (const float*)d_in[0];   // (NV, NB, 2) f32
    const int*   inds = (const int*)d_in[1];     // (NV, NB)    i32
    const float* ft   = (const float*)d_in[2];   // (NUM_ATLAS, FD, FS, FS) f32
    float*       out  = (float*)d_out;           // (NV, NB, FD) f32

    const int nsamp  = in_sizes[1];              // NV*NB = 524288
    const int blocks = nsamp / (16 * 8);         // 16 samples/wave, 8 waves/block (exact: 4096)
    featex_wmma_kernel<<<blocks, 256, 0, stream>>>(x, inds, ft, out, nsamp);
}